// GeometricController_37366215475654
// MI455X (gfx1250) — compile-verified
//
#include <hip/hip_runtime.h>

// Single-wave geometric controller for MI455X (gfx1250, wave32).
// Phase 1: replicated scalar chain (VALU). Phase 2: one V_WMMA_F32_16X16X4_F32
// batching the five 3x3 products Rd^T * {Rd_d, Rd_dd, Rd_ddd, Rd_dddd, R}.
// Phase 3: skew-algebra closed forms + output.

typedef __attribute__((ext_vector_type(2))) float v2f;
typedef __attribute__((ext_vector_type(8))) float v8f;

#define KP_   16.0f
#define KV_   5.6f
#define KORI_ 8.81f
#define KW_   2.54f
#define MASS_ 4.34f
#define G_    9.81f

struct V3 { float x, y, z; };

__device__ __forceinline__ V3 mkv(float x, float y, float z) { V3 r; r.x = x; r.y = y; r.z = z; return r; }
__device__ __forceinline__ V3 operator+(V3 a, V3 b) { return mkv(a.x + b.x, a.y + b.y, a.z + b.z); }
__device__ __forceinline__ V3 operator-(V3 a, V3 b) { return mkv(a.x - b.x, a.y - b.y, a.z - b.z); }
__device__ __forceinline__ V3 operator*(float s, V3 a) { return mkv(s * a.x, s * a.y, s * a.z); }
__device__ __forceinline__ V3 operator-(V3 a) { return mkv(-a.x, -a.y, -a.z); }
__device__ __forceinline__ float dot3(V3 a, V3 b) { return a.x * b.x + a.y * b.y + a.z * b.z; }
__device__ __forceinline__ V3 cross3(V3 a, V3 b) {
  return mkv(a.y * b.z - a.z * b.y, a.z * b.x - a.x * b.z, a.x * b.y - a.y * b.x);
}
__device__ __forceinline__ V3 unit(V3 a) {           // v / |v|
  float inv = 1.0f / sqrtf(dot3(a, a));
  return inv * a;
}
__device__ __forceinline__ V3 unitneg(V3 a) {        // -v / |v|
  float inv = 1.0f / sqrtf(dot3(a, a));
  return (-inv) * a;
}
__device__ __forceinline__ V3 ld3(const float* p, int i) { return mkv(p[3 * i], p[3 * i + 1], p[3 * i + 2]); }
// rows r0,r1,r2:  M * v
__device__ __forceinline__ V3 matv(V3 r0, V3 r1, V3 r2, V3 v) {
  return mkv(dot3(r0, v), dot3(r1, v), dot3(r2, v));
}
// rows r0,r1,r2:  M^T * v
__device__ __forceinline__ V3 matTv(V3 r0, V3 r1, V3 r2, V3 v) {
  return mkv(r0.x * v.x + r1.x * v.y + r2.x * v.z,
             r0.y * v.x + r1.y * v.y + r2.y * v.z,
             r0.z * v.x + r1.z * v.y + r2.z * v.z);
}

__global__ __launch_bounds__(32) void GeometricController_kernel(
    const float* __restrict__ st, const float* __restrict__ rs,
    const float* __restrict__ Jm, float* __restrict__ out) {
  __shared__ float ldsIn[6 * 9];   // mats: 0=Rd 1=Rd_d 2=Rd_dd 3=Rd_ddd 4=Rd_dddd 5=R  (row-major)
  __shared__ float ldsOut[5 * 9];  // P1..P4 = Rd^T Rd_*, P5 = Rd^T R

  const int lane = threadIdx.x;

  // ---------------- inputs (uniform addresses -> SMEM-friendly) ----------------
  V3 pos = mkv(st[0], st[1], st[2]);
  float qx = st[3], qy = st[4], qz = st[5], qw = st[6];
  V3 vel = mkv(st[7], st[8], st[9]);
  V3 w   = mkv(st[10], st[11], st[12]);

  V3 des_pos      = ld3(rs, 0),  des_vel     = ld3(rs, 1);
  V3 des_acc      = ld3(rs, 2),  des_acc_d   = ld3(rs, 3);
  V3 des_acc_dd   = ld3(rs, 4),  des_acc_ddd = ld3(rs, 5);
  V3 des_acc_dddd = ld3(rs, 6);
  V3 des_b1       = ld3(rs, 7),  des_b1_d    = ld3(rs, 8);
  V3 des_b1_dd    = ld3(rs, 9),  des_b1_ddd  = ld3(rs, 10);

  V3 j0 = mkv(Jm[0], Jm[1], Jm[2]);
  V3 j1 = mkv(Jm[3], Jm[4], Jm[5]);
  V3 j2 = mkv(Jm[6], Jm[7], Jm[8]);

  // ---------------- R from quaternion (x,y,z,w) ----------------
  V3 r0 = mkv(1.0f - 2.0f * (qy * qy + qz * qz), 2.0f * (qx * qy - qz * qw), 2.0f * (qx * qz + qy * qw));
  V3 r1 = mkv(2.0f * (qx * qy + qz * qw), 1.0f - 2.0f * (qx * qx + qz * qz), 2.0f * (qy * qz - qx * qw));
  V3 r2 = mkv(2.0f * (qx * qz - qy * qw), 2.0f * (qy * qz + qx * qw), 1.0f - 2.0f * (qx * qx + qy * qy));

  V3 e3 = mkv(0.0f, 0.0f, 1.0f);
  V3 b3 = mkv(r0.z, r1.z, r2.z);                 // R e3
  V3 u1 = mkv(w.y, -w.x, 0.0f);                  // w x e3
  V3 u2 = cross3(w, u1);
  V3 u3 = cross3(w, u2);
  V3 b3_d   = matv(r0, r1, r2, u1);
  V3 b3_dd  = matv(r0, r1, r2, u2);
  V3 b3_ddd = matv(r0, r1, r2, u3);

  // ---------------- serial thrust / des_b3 chain ----------------
  V3 des_b3 = -(KP_ * (pos - des_pos) + KV_ * (vel - des_vel)) - (MASS_ * G_) * e3 + MASS_ * des_acc;
  float thrust = -dot3(des_b3, b3);

  V3 ev_d = G_ * e3 - (thrust / MASS_) * b3 - des_acc;
  V3 des_b3_d = -(KP_ * (vel - des_vel)) - KV_ * ev_d + MASS_ * des_acc_d;

  float thrust_d = -dot3(des_b3_d, b3) - dot3(des_b3, b3_d);
  V3 ev_dd = (1.0f / MASS_) * (-(thrust_d)*b3 - thrust * b3_d) - des_acc_d;
  V3 des_b3_dd = -(KP_ * ev_d) - KV_ * ev_dd + MASS_ * des_acc_dd;

  float thrust_dd = -dot3(des_b3_dd, b3) - 2.0f * dot3(des_b3_d, b3_d) - dot3(des_b3, b3_dd);
  V3 ev_ddd = (1.0f / MASS_) * (-(thrust_dd)*b3 - 2.0f * thrust_d * b3_d - thrust * b3_dd) - des_acc_ddd;
  V3 des_b3_ddd = -(KP_ * ev_dd) - KV_ * ev_ddd + MASS_ * des_acc_ddd;

  float thrust_ddd = -dot3(des_b3_ddd, b3) - 3.0f * dot3(des_b3_dd, b3_d)
                     - 3.0f * dot3(des_b3_d, b3_dd) - dot3(des_b3, b3_ddd);
  V3 ev_dddd = (1.0f / MASS_) * (-(thrust_ddd)*b3 - 3.0f * thrust_dd * b3_d
                                 - 3.0f * thrust_d * b3_dd - thrust * b3_ddd) - des_acc_dddd;
  V3 des_b3_dddd = -(KP_ * ev_ddd) - KV_ * ev_dddd + MASS_ * des_acc_dddd;

  // normalized, negated des_b3 series
  V3 d3   = unitneg(des_b3);
  V3 d3d  = unitneg(des_b3_d);
  V3 d3dd = unitneg(des_b3_dd);
  V3 d3d3 = unitneg(des_b3_ddd);
  V3 d3d4 = unitneg(des_b3_dddd);

  // b2 series
  V3 b2   = cross3(d3, des_b1);
  V3 b2d  = cross3(d3d, des_b1) + cross3(d3, des_b1_d);
  V3 b2dd = cross3(d3dd, des_b1) + 2.0f * cross3(d3d, des_b1_d) + cross3(d3, des_b1_dd);
  V3 b2d3 = cross3(d3d3, des_b1) + 3.0f * cross3(d3dd, des_b1_d)
          + 3.0f * cross3(d3d, des_b1_dd) + cross3(d3, des_b1_ddd);
  V3 b2d4 = cross3(d3d4, des_b1) + 4.0f * cross3(d3d3, des_b1_d)
          + 6.0f * cross3(d3dd, des_b1_dd) + 4.0f * cross3(d3d, des_b1_ddd);
  b2 = unit(b2); b2d = unit(b2d); b2dd = unit(b2dd); b2d3 = unit(b2d3); b2d4 = unit(b2d4);

  // b1 series
  V3 b1   = cross3(b2, d3);
  V3 b1d  = cross3(b2d, d3) + cross3(b2, d3d);
  V3 b1dd = cross3(b2dd, d3) + 2.0f * cross3(b2d, d3d) + cross3(b2, d3dd);
  V3 b1d3 = cross3(b2d3, d3) + 3.0f * cross3(b2dd, d3d) + 3.0f * cross3(b2d, d3dd) + cross3(b2, d3d3);
  V3 b1d4 = cross3(b2d4, d3) + 4.0f * cross3(b2d3, d3d) + 6.0f * cross3(b2dd, d3dd)
          + 4.0f * cross3(b2d, d3d3) + cross3(b2, d3d4);
  b1 = unit(b1); b1d = unit(b1d); b1dd = unit(b1dd); b1d3 = unit(b1d3); b1d4 = unit(b1d4);

  // ---------------- stage matrices in LDS (row-major; columns = b1,b2,d3) ----------------
  if (lane == 0) {
    float* p;
    p = &ldsIn[0 * 9]; p[0]=b1.x;  p[1]=b2.x;  p[2]=d3.x;  p[3]=b1.y;  p[4]=b2.y;  p[5]=d3.y;  p[6]=b1.z;  p[7]=b2.z;  p[8]=d3.z;   // Rd
    p = &ldsIn[1 * 9]; p[0]=b1d.x; p[1]=b2d.x; p[2]=d3d.x; p[3]=b1d.y; p[4]=b2d.y; p[5]=d3d.y; p[6]=b1d.z; p[7]=b2d.z; p[8]=d3d.z;  // Rd_d
    p = &ldsIn[2 * 9]; p[0]=b1dd.x;p[1]=b2dd.x;p[2]=d3dd.x;p[3]=b1dd.y;p[4]=b2dd.y;p[5]=d3dd.y;p[6]=b1dd.z;p[7]=b2dd.z;p[8]=d3dd.z; // Rd_dd
    p = &ldsIn[3 * 9]; p[0]=b1d3.x;p[1]=b2d3.x;p[2]=d3d3.x;p[3]=b1d3.y;p[4]=b2d3.y;p[5]=d3d3.y;p[6]=b1d3.z;p[7]=b2d3.z;p[8]=d3d3.z; // Rd_ddd
    p = &ldsIn[4 * 9]; p[0]=b1d4.x;p[1]=b2d4.x;p[2]=d3d4.x;p[3]=b1d4.y;p[4]=b2d4.y;p[5]=d3d4.y;p[6]=b1d4.z;p[7]=b2d4.z;p[8]=d3d4.z; // Rd_dddd
    p = &ldsIn[5 * 9]; p[0]=r0.x;  p[1]=r0.y;  p[2]=r0.z;  p[3]=r1.x;  p[4]=r1.y;  p[5]=r1.z;  p[6]=r2.x;  p[7]=r2.y;  p[8]=r2.z;   // R
  }
  __syncthreads();

  // ---------------- WMMA fragments (V_WMMA_F32_16X16X4_F32) ----------------
  // A = Rd^T (16x4, rows 0..2 live):  A[m][k] = Rd[k][m]
  const int m  = lane & 15;
  const int hi = lane >> 4;  // 0: K={0,1}, 1: K={2,3}
  float ax = (m < 3) ? ldsIn[6 * hi + m] : 0.0f;        // Rd[0][m] or Rd[2][m]
  float ay = (m < 3 && hi == 0) ? ldsIn[3 + m] : 0.0f;  // Rd[1][m]
  // B = [Rd_d | Rd_dd | Rd_ddd | Rd_dddd | R] as 4x16 (rows 0..2 live, cols 0..14 live)
  int  msel  = m / 3; if (msel > 4) msel = 4;
  const int  base  = (1 + msel) * 9 + (m % 3);
  const bool valid = (m < 15);
  float bx = valid ? ldsIn[base + 6 * hi] : 0.0f;              // X[0][c] or X[2][c]
  float by = (valid && hi == 0) ? ldsIn[base + 3] : 0.0f;      // X[1][c]

  v2f a; a[0] = ax; a[1] = ay;
  v2f b; b[0] = bx; b[1] = by;
  v8f c = {0.0f, 0.0f, 0.0f, 0.0f, 0.0f, 0.0f, 0.0f, 0.0f};
  // D = A * B + 0 : five 3x3 products in one FP32 matrix op
  c = __builtin_amdgcn_wmma_f32_16x16x4_f32(false, a, false, b, (short)0, c, false, false);

  // D rows 0..2 live in c[0..2] of lanes 0..14 (C/D layout: lane=N, VGPR=M)
  if (lane < 15) {
    const int mat = lane / 3, col = lane % 3;
    ldsOut[mat * 9 + 0 + col] = c[0];
    ldsOut[mat * 9 + 3 + col] = c[1];
    ldsOut[mat * 9 + 6 + col] = c[2];
  }
  __syncthreads();

#define PM(i, r, cc) ldsOut[(i) * 9 + (r) * 3 + (cc)]
  // w_d = skew2vec(P1)
  V3 w_d = mkv(-PM(0, 1, 2), PM(0, 0, 2), -PM(0, 0, 1));
  // a_d = skew2vec(P2 - W^2);   (W^2)[i][j]_offdiag = u_i u_j
  V3 a_d = mkv(-(PM(1, 1, 2) - w_d.y * w_d.z),
                 PM(1, 0, 2) - w_d.x * w_d.z,
               -(PM(1, 0, 1) - w_d.x * w_d.y));
  // j_d = skew2vec(P3 - 3 W skew(a_d));  (W*Sa)[i][j]_offdiag = a_i u_j
  V3 j_d = mkv(-(PM(2, 1, 2) - 3.0f * a_d.y * w_d.z),
                 PM(2, 0, 2) - 3.0f * a_d.x * w_d.z,
               -(PM(2, 0, 1) - 3.0f * a_d.x * w_d.y));
  // s_d = skew2vec(P4 - 4 W skew(j_d) - 3 skew(a_d)^2)
  V3 s_d = mkv(-(PM(3, 1, 2) - 4.0f * j_d.y * w_d.z - 3.0f * a_d.y * a_d.z),
                 PM(3, 0, 2) - 4.0f * j_d.x * w_d.z - 3.0f * a_d.x * a_d.z,
               -(PM(3, 0, 1) - 4.0f * j_d.x * w_d.y - 3.0f * a_d.x * a_d.y));
  // e_R = 0.5 skew2vec(P5 - P5^T)   with P5 = Rd^T R  (R^T Rd = P5^T)
  V3 e_R = 0.5f * mkv(-(PM(4, 1, 2) - PM(4, 2, 1)),
                        PM(4, 0, 2) - PM(4, 2, 0),
                      -(PM(4, 0, 1) - PM(4, 1, 0)));
#undef PM

  // Rd * v via columns, then R^T
  V3 Rdwd  = w_d.x * b1 + w_d.y * b2 + w_d.z * d3;
  V3 e_w   = w - matTv(r0, r1, r2, Rdwd);

  V3 t     = w_d - a_d - j_d - s_d;
  V3 Rdt   = t.x * b1 + t.y * b2 + t.z * d3;
  V3 tmpM  = cross3(w, matTv(r0, r1, r2, Rdt));

  V3 Jw    = matv(j0, j1, j2, w);
  V3 M     = -(KORI_ * e_R) - KW_ * e_w + cross3(w, Jw) - matv(j0, j1, j2, tmpM);

  if (lane == 0) {
    out[0] = fmaxf(0.0f, thrust);
    out[1] = M.x;
    out[2] = M.y;
    out[3] = M.z;
  }
}

extern "C" void kernel_launch(void* const* d_in, const int* in_sizes, int n_in,
                              void* d_out, int out_size, void* d_ws, size_t ws_size,
                              hipStream_t stream) {
  (void)in_sizes; (void)n_in; (void)out_size; (void)d_ws; (void)ws_size;
  const float* state = (const float*)d_in[0];
  const float* ref   = (const float*)d_in[1];
  const float* J     = (const float*)d_in[2];
  float* out = (float*)d_out;
  GeometricController_kernel<<<1, 32, 0, stream>>>(state, ref, J, out);
}